// BlockLearnableCompressionMatrix_59021440582053
// MI455X (gfx1250) — compile-verified
//
#include <hip/hip_runtime.h>
#include <hip/hip_bf16.h>
#include <math.h>

// Shapes from the reference:
//   x      : (16, 1, 1024, 2048) f32
//   blocks : (128, 8) f32
//   out    : Ax (16,128,2048) f32  ++  A_full (128,1024) f32
#define BSZ     16
#define NCH     1024
#define TLEN    2048
#define MOUT    128
#define CBLK    8

typedef __attribute__((ext_vector_type(2))) float v2f;
typedef __attribute__((ext_vector_type(8))) float v8f;

// ---------------------------------------------------------------------------
// Kernel 1: A_full = blockdiag(tanh(blocks))  -> 128 x 1024
// ---------------------------------------------------------------------------
__global__ void blcm_afull_kernel(const float* __restrict__ blocks,
                                  float* __restrict__ afull) {
    int idx = blockIdx.x * blockDim.x + threadIdx.x;   // 0 .. 131071
    if (idx < MOUT * NCH) {
        int m = idx >> 10;        // row
        int n = idx & 1023;       // col
        float v = 0.0f;
        if ((n >> 3) == m) v = tanhf(blocks[m * CBLK + (n & 7)]);
        afull[idx] = v;
    }
}

// ---------------------------------------------------------------------------
// Kernel 2: Ax[b, m0+M, t0+N] via V_WMMA_F32_16X16X4_F32.
// One wave computes a 16(m) x 16(t) tile = GEMM 16x16x128 over the dense
// channel window [8*m0, 8*m0+128), split into 32 K=4 chunks.
//
// A (16x4 f32, 2 VGPRs): lanes 0-15 K={0,1}, lanes 16-31 K={2,3}; row M = lane%16.
//   Row mrow is nonzero only in chunks ka=2*mrow (block cols {2h,2h+1}) and
//   kb=2*mrow+1 (block cols {4+2h,5+2h}), h = lane>=16.
// B (4x16 f32, 2 VGPRs): VGPR0/1 = K rows {0,1} (lo lanes) / {2,3} (hi lanes),
//   N = t0 + lane%16  -> 64B coalesced per half-wave per row.
// D (16x16 f32, 8 VGPRs): lanes 0-15 M=v, lanes 16-31 M=v+8.
// ---------------------------------------------------------------------------
__global__ void __launch_bounds__(256)
blcm_ax_wmma_kernel(const float* __restrict__ x,
                    const float* __restrict__ blocks,
                    float* __restrict__ out_ax) {
    const int bid  = blockIdx.x;          // 0 .. 2047
    const int tg   = bid & 15;            // 16 groups of 8 t-tiles
    const int mt   = (bid >> 4) & 7;      // 8 m-tiles of 16 rows
    const int b    = bid >> 7;            // 16 batches
    const int wave = threadIdx.x >> 5;    // 8 waves / block
    const int lane = threadIdx.x & 31;

    const int mrow = lane & 15;           // M row (or N col) within tile
    const int hi   = lane >> 4;           // 0: lanes 0-15, 1: lanes 16-31

    const int m0 = mt * 16;               // first m of this tile
    const int t0 = (tg * 8 + wave) * 16;  // first t of this tile
    const int gm = m0 + mrow;             // global m for this lane's A row

    // Per-lane weights: tanh of the 4 block entries this lane contributes.
    const float* wrow = blocks + (size_t)gm * CBLK;
    const float wa0 = tanhf(wrow[2 * hi + 0]);   // chunk ka, K pair
    const float wa1 = tanhf(wrow[2 * hi + 1]);
    const float wb0 = tanhf(wrow[4 + 2 * hi]);   // chunk kb, K pair
    const float wb1 = tanhf(wrow[5 + 2 * hi]);
    const int ka = 2 * mrow;
    const int kb = 2 * mrow + 1;

    // x base for this lane's B column: channel window start + t column.
    const float* xb = x + (size_t)b * (NCH * TLEN)
                        + (size_t)(m0 * CBLK) * TLEN   // channel window start
                        + t0 + mrow;                    // N = lane%16
    const size_t hioff = (size_t)(2 * hi) * TLEN;       // hi lanes: K rows {2,3}

    v8f acc = {};
#pragma unroll
    for (int k = 0; k < 32; ++k) {
        // B chunk: rows 4k+2h, 4k+2h+1 of the 128-deep channel window.
        const float* p = xb + (size_t)(4 * k) * TLEN + hioff;
        v2f Bv;
        Bv.x = p[0];
        Bv.y = p[TLEN];

        // A chunk: nonzero only for this lane's two owned chunks.
        v2f Av;
        Av.x = (k == ka) ? wa0 : ((k == kb) ? wb0 : 0.0f);
        Av.y = (k == ka) ? wa1 : ((k == kb) ? wb1 : 0.0f);

        // D = A x B + C   (fp32 WMMA, exact vs reference accumulation class)
        acc = __builtin_amdgcn_wmma_f32_16x16x4_f32(
            /*neg_a=*/false, Av, /*neg_b=*/false, Bv,
            /*c_mod=*/(short)0, acc, /*reuse_a=*/false, /*reuse_b=*/false);
    }

    // Store D: element v -> row M = v + 8*hi, col N = mrow (t0 + mrow).
    float* o = out_ax + ((size_t)b * MOUT + m0) * TLEN + t0 + mrow;
#pragma unroll
    for (int v = 0; v < 8; ++v) {
        o[(size_t)(v + 8 * hi) * TLEN] = acc[v];
    }
}

// ---------------------------------------------------------------------------
extern "C" void kernel_launch(void* const* d_in, const int* in_sizes, int n_in,
                              void* d_out, int out_size, void* d_ws, size_t ws_size,
                              hipStream_t stream) {
    const float* x      = (const float*)d_in[0];
    const float* blocks = (const float*)d_in[1];
    float* out_ax    = (float*)d_out;                                  // 16*128*2048
    float* out_afull = (float*)d_out + (size_t)BSZ * MOUT * TLEN;      // 128*1024

    // A_full: 131072 elements
    blcm_afull_kernel<<<(MOUT * NCH + 255) / 256, 256, 0, stream>>>(blocks, out_afull);

    // Ax: 16 b * 8 m-tiles * 16 t-groups = 2048 blocks, 8 waves each.
    blcm_ax_wmma_kernel<<<BSZ * 8 * 16, 256, 0, stream>>>(x, blocks, out_ax);
}